// GCN_4990751998361
// MI455X (gfx1250) — compile-verified
//
#include <hip/hip_runtime.h>
#include <math.h>

#define H   64
#define NG  128
#define NC  5

typedef __attribute__((ext_vector_type(2))) float v2f;
typedef __attribute__((ext_vector_type(8))) float v8f;

// ---------------- degree / norm ----------------
__global__ void k_deg_init(float* deg, int N) {
  int i = blockIdx.x * blockDim.x + threadIdx.x;
  if (i < N) deg[i] = 1.0f;                      // self-loop contributes 1
}

__global__ void k_deg_scatter(const int* __restrict__ ei, int E, float* deg) {
  int e = blockIdx.x * blockDim.x + threadIdx.x;
  if (e < E) atomicAdd(&deg[ei[E + e]], 1.0f);   // dst row is ei[1,:]
}

__global__ void k_dinv(float* deg, int N) {
  int i = blockIdx.x * blockDim.x + threadIdx.x;
  if (i < N) deg[i] = rsqrtf(deg[i]);            // deg >= 1 always
}

// ---------------- layer-1 GEMM (K=2, trivial) ----------------
__global__ void k_gemm_in(const float* __restrict__ x, const float* __restrict__ W1,
                          float* __restrict__ t, int N) {
  int i = blockIdx.x * blockDim.x + threadIdx.x;
  if (i < N * H) {
    int n = i / H, j = i - n * H;
    t[i] = fmaf(x[2 * n], W1[j], x[2 * n + 1] * W1[H + j]);
  }
}

// ---------------- aggregation: init (bias + self-loop term) ----------------
__global__ void k_agg_init(const float* __restrict__ t, const float* __restrict__ dinv,
                           const float* __restrict__ b, float* __restrict__ out, int N) {
  int i = blockIdx.x * blockDim.x + threadIdx.x;
  if (i < N * H) {
    int n = i / H, j = i - n * H;
    float di = dinv[n];
    out[i] = fmaf(di * di, t[i], b[j]);
  }
}

// ---------------- aggregation: edge scatter (float4 gather + 4 atomics) ----------------
__global__ void k_agg_edges(const int* __restrict__ ei, int E,
                            const float* __restrict__ t, const float* __restrict__ dinv,
                            float* __restrict__ out) {
  int tid = blockIdx.x * blockDim.x + threadIdx.x;   // E*16 threads
  int e = tid >> 4;
  int c = tid & 15;
  if (e < E) {
    int s = ei[e];
    int d = ei[E + e];
    float nrm = dinv[s] * dinv[d];
    float4 v = *(const float4*)(t + (size_t)s * H + c * 4);
    float* op = out + (size_t)d * H + c * 4;
    atomicAdd(op + 0, nrm * v.x);
    atomicAdd(op + 1, nrm * v.y);
    atomicAdd(op + 2, nrm * v.z);
    atomicAdd(op + 3, nrm * v.w);
  }
}

// ---------------- WMMA fp32 GEMM: t = relu(hIn) @ W, [N,64]x[64,64] ----------------
// One wave computes a 16x16 output tile; 4 waves/block cover all 64 columns of a
// 16-row strip. K=64 consumed in 16 steps of V_WMMA_F32_16X16X4_F32.
// Loads are UNGUARDED: grid = ceil(N/16) reads at most 15 rows (3.84 KB) past the
// live region, which stays inside d_ws (>=33 KB of workspace follows each buffer);
// tail-row garbage only affects D rows that are never stored. This keeps EXEC
// all-ones and branch-free through the WMMA sequence.
__global__ void k_gemm_wmma(const float* __restrict__ hIn, const float* __restrict__ W,
                            float* __restrict__ tOut, int N) {
  int lane = threadIdx.x & 31;
  int wave = threadIdx.x >> 5;           // column tile 0..3
  int rowBase = blockIdx.x * 16;
  int colBase = wave * 16;
  int m  = lane & 15;                    // M index for A, N index for B
  int kp = (lane >> 4) << 1;             // 0 or 2 within the K=4 slice
  int row = rowBase + m;
  int col = colBase + m;

  const float* aRow = hIn + (size_t)row * H + kp;   // 8-byte aligned (kp even)
  const float* bCol = W + (size_t)kp * H + col;

  v8f acc = {};
  #pragma unroll
  for (int k0 = 0; k0 < H; k0 += 4) {
    v2f a = *(const v2f*)(aRow + k0);               // global_load_b64
    a.x = fmaxf(a.x, 0.0f);
    a.y = fmaxf(a.y, 0.0f);
    v2f b;
    b.x = bCol[(size_t)k0 * H];
    b.y = bCol[(size_t)k0 * H + H];
    acc = __builtin_amdgcn_wmma_f32_16x16x4_f32(
        /*neg_a=*/false, a, /*neg_b=*/false, b,
        /*c_mod=*/(short)0, acc, /*reuse_a=*/false, /*reuse_b=*/false);
  }

  // D layout: lanes 0-15 hold M=r, lanes 16-31 hold M=8+r; N = lane&15
  int rbase = rowBase + ((lane >> 4) << 3);
  int ocol = colBase + (lane & 15);
  #pragma unroll
  for (int r = 0; r < 8; r++) {
    int rr = rbase + r;
    if (rr < N) tOut[(size_t)rr * H + ocol] = acc[r];
  }
}

// ---------------- pooling ----------------
__global__ void k_pool_init(float* sums, float* cnts) {
  int i = blockIdx.x * blockDim.x + threadIdx.x;
  if (i < NG * H) sums[i] = 0.0f;
  if (i < NG)     cnts[i] = 0.0f;
}

__global__ void k_pool_scatter(const float* __restrict__ h3, const int* __restrict__ batch,
                               float* sums, float* cnts, int N) {
  int tid = blockIdx.x * blockDim.x + threadIdx.x;   // N*16 threads
  int n = tid >> 4;
  int c = tid & 15;
  if (n < N) {
    int g = batch[n];
    float4 v = *(const float4*)(h3 + (size_t)n * H + c * 4);
    float* sp = sums + (size_t)g * H + c * 4;
    atomicAdd(sp + 0, v.x);
    atomicAdd(sp + 1, v.y);
    atomicAdd(sp + 2, v.z);
    atomicAdd(sp + 3, v.w);
    if (c == 0) atomicAdd(&cnts[g], 1.0f);
  }
}

// ---------------- head: concat-GEMM + log_softmax ----------------
__global__ void k_head(const float* __restrict__ sums, const float* __restrict__ cnts,
                       const float* __restrict__ ge, const float* __restrict__ Wl,
                       const float* __restrict__ bl, float* __restrict__ out) {
  int g = threadIdx.x;
  if (g >= NG) return;
  float inv = 1.0f / fmaxf(cnts[g], 1.0f);
  float z[NC];
  #pragma unroll
  for (int c = 0; c < NC; c++) z[c] = bl[c];
  for (int j = 0; j < H; j++) {
    float p = sums[(size_t)g * H + j] * inv;
    #pragma unroll
    for (int c = 0; c < NC; c++) z[c] = fmaf(p, Wl[j * NC + c], z[c]);
  }
  for (int j = 0; j < H; j++) {
    float e = ge[(size_t)g * H + j];
    #pragma unroll
    for (int c = 0; c < NC; c++) z[c] = fmaf(e, Wl[(H + j) * NC + c], z[c]);
  }
  float mx = z[0];
  #pragma unroll
  for (int c = 1; c < NC; c++) mx = fmaxf(mx, z[c]);
  float se = 0.0f;
  #pragma unroll
  for (int c = 0; c < NC; c++) se += expf(z[c] - mx);
  float lse = mx + logf(se);
  #pragma unroll
  for (int c = 0; c < NC; c++) out[g * NC + c] = z[c] - lse;
}

static inline int cdiv(long long a, long long b) { return (int)((a + b - 1) / b); }

extern "C" void kernel_launch(void* const* d_in, const int* in_sizes, int n_in,
                              void* d_out, int out_size, void* d_ws, size_t ws_size,
                              hipStream_t stream) {
  const float* x     = (const float*)d_in[0];
  const int*   ei    = (const int*)  d_in[1];
  const int*   batch = (const int*)  d_in[2];
  const float* ge    = (const float*)d_in[3];
  const float* W1 = (const float*)d_in[4];  const float* b1 = (const float*)d_in[5];
  const float* W2 = (const float*)d_in[6];  const float* b2 = (const float*)d_in[7];
  const float* W3 = (const float*)d_in[8];  const float* b3 = (const float*)d_in[9];
  const float* Wl = (const float*)d_in[10]; const float* bl = (const float*)d_in[11];
  float* out = (float*)d_out;

  int N = in_sizes[0] / 2;   // x is [N,2]
  int E = in_sizes[1] / 2;   // edge_index is [2,E]

  // workspace layout (floats): dinv[N] | bufA[N*H] | bufB[N*H] | sums[NG*H] | cnts[NG]
  float* ws   = (float*)d_ws;
  float* dinv = ws;
  float* bufA = ws + (((size_t)N + 255) & ~(size_t)255);
  float* bufB = bufA + (size_t)N * H;
  float* sums = bufB + (size_t)N * H;
  float* cnts = sums + (size_t)NG * H;

  // degrees -> dinv
  k_deg_init   <<<cdiv(N, 256), 256, 0, stream>>>(dinv, N);
  k_deg_scatter<<<cdiv(E, 256), 256, 0, stream>>>(ei, E, dinv);
  k_dinv       <<<cdiv(N, 256), 256, 0, stream>>>(dinv, N);

  // layer 1: bufA = x@W1 ; bufB = agg(bufA) + b1
  k_gemm_in  <<<cdiv((long long)N * H, 256), 256, 0, stream>>>(x, W1, bufA, N);
  k_agg_init <<<cdiv((long long)N * H, 256), 256, 0, stream>>>(bufA, dinv, b1, bufB, N);
  k_agg_edges<<<cdiv((long long)E * 16, 256), 256, 0, stream>>>(ei, E, bufA, dinv, bufB);

  // layer 2: bufA = relu(bufB)@W2 ; bufB = agg(bufA) + b2
  k_gemm_wmma<<<cdiv(N, 16), 128, 0, stream>>>(bufB, W2, bufA, N);
  k_agg_init <<<cdiv((long long)N * H, 256), 256, 0, stream>>>(bufA, dinv, b2, bufB, N);
  k_agg_edges<<<cdiv((long long)E * 16, 256), 256, 0, stream>>>(ei, E, bufA, dinv, bufB);

  // layer 3: bufA = relu(bufB)@W3 ; bufB = agg(bufA) + b3
  k_gemm_wmma<<<cdiv(N, 16), 128, 0, stream>>>(bufB, W3, bufA, N);
  k_agg_init <<<cdiv((long long)N * H, 256), 256, 0, stream>>>(bufA, dinv, b3, bufB, N);
  k_agg_edges<<<cdiv((long long)E * 16, 256), 256, 0, stream>>>(ei, E, bufA, dinv, bufB);

  // pooling + head
  k_pool_init   <<<cdiv(NG * H, 256), 256, 0, stream>>>(sums, cnts);
  k_pool_scatter<<<cdiv((long long)N * 16, 256), 256, 0, stream>>>(bufB, batch, sums, cnts, N);
  k_head<<<1, NG, 0, stream>>>(sums, cnts, ge, Wl, bl, out);
}